// HungarianMatcher_5952824672675
// MI455X (gfx1250) — compile-verified
//
#include <hip/hip_runtime.h>
#include <stdint.h>

#define NT   1600   // targets
#define NC   91     // classes
#define QPB  8      // queries (waves) per block
#define TPB  256    // 8 waves of 32 (wave32 on gfx1250)

// CDNA5 async global->LDS copy (GVS mode: SGPR base + per-lane VGPR offset).
// Tracked with ASYNCcnt; completion via s_wait_asynccnt.
__device__ __forceinline__ void async_load_b128(uint32_t lds_off, uint32_t goff,
                                                const void* base) {
  asm volatile("global_load_async_to_lds_b128 %0, %1, %2"
               :: "v"(lds_off), "v"(goff), "s"(base)
               : "memory");
}

__global__ __launch_bounds__(TPB) void detr_cost_kernel(
    const float* __restrict__ logits,    // [BQ, 91]
    const float* __restrict__ qboxes,    // [BQ, 4] cxcywh
    const float* __restrict__ tboxes,    // [NT, 4] cxcywh
    const int*   __restrict__ tlabels,   // [NT]
    float* __restrict__ out,             // [BQ, NT]
    int bq_total)
{
  __shared__ float4 tb4[NT];         // staged cxcywh, converted in-place to xyxy
  __shared__ int    tlab[NT];        // staged labels (raw)
  __shared__ float2 taux[NT];        // .x = area, .y = bitcast(label)
  __shared__ float  probs[QPB][92];

  const int tid = threadIdx.x;

  // ---- Stage target boxes (25600 B = 1600 x b128) and labels (6400 B =
  // ---- 400 x b128) into LDS via CDNA5 async loads; straight-line issue.
  {
    const uint32_t ldsA = (uint32_t)(uintptr_t)(void*)&tb4[0];
    const uint32_t ldsL = (uint32_t)(uintptr_t)(void*)&tlab[0];
    uint32_t o = (uint32_t)tid * 16u;
#pragma unroll
    for (int r = 0; r < 6; ++r) {                 // 6*256 = 1536 chunks
      async_load_b128(ldsA + o + r * 4096u, o + r * 4096u, (const void*)tboxes);
    }
    if (tid < 64)                                 // chunks 1536..1599
      async_load_b128(ldsA + o + 24576u, o + 24576u, (const void*)tboxes);
    async_load_b128(ldsL + o, o, (const void*)tlabels);        // chunks 0..255
    if (tid < 144)                                // chunks 256..399
      async_load_b128(ldsL + o + 4096u, o + 4096u, (const void*)tlabels);
    asm volatile("s_wait_asynccnt 0x0" ::: "memory");
  }
  __syncthreads();

  // ---- Per-target precompute: cxcywh -> xyxy (in place), pack area+label.
  for (int j = tid; j < NT; j += TPB) {
    float4 b = tb4[j];
    float x0 = b.x - 0.5f * b.z, y0 = b.y - 0.5f * b.w;
    float x1 = b.x + 0.5f * b.z, y1 = b.y + 0.5f * b.w;
    tb4[j] = make_float4(x0, y0, x1, y1);
    taux[j] = make_float2((x1 - x0) * (y1 - y0), __int_as_float(tlab[j]));
  }
  __syncthreads();

  const int wave = tid >> 5;
  const int lane = tid & 31;
  const int q = blockIdx.x * QPB + wave;
  if (q >= bq_total) return;   // no barriers beyond this point

  // ---- Wave-parallel softmax over 91 classes (lanes cover c, c+32, c+64).
  const float* lrow = logits + (size_t)q * NC;
  float a0 = lrow[lane];
  float a1 = lrow[lane + 32];
  float a2 = (lane + 64 < NC) ? lrow[lane + 64] : -3.4e38f;
  float m = fmaxf(a0, fmaxf(a1, a2));
  for (int d = 16; d; d >>= 1) m = fmaxf(m, __shfl_xor(m, d, 32));
  float e0 = __expf(a0 - m);
  float e1 = __expf(a1 - m);
  float e2 = (lane + 64 < NC) ? __expf(a2 - m) : 0.0f;
  float s = e0 + e1 + e2;
  for (int d = 16; d; d >>= 1) s += __shfl_xor(s, d, 32);
  float inv = __builtin_amdgcn_rcpf(s);
  probs[wave][lane]      = e0 * inv;
  probs[wave][lane + 32] = e1 * inv;
  if (lane + 64 < NC) probs[wave][lane + 64] = e2 * inv;
  // probs row is wave-private; same-wave LDS ops are in-order -> no barrier.

  // ---- Per-query constants (uniform across the wave).
  const float* qb = qboxes + (size_t)q * 4;
  float qcx = qb[0], qcy = qb[1], qw = qb[2], qh = qb[3];
  float qx0 = qcx - 0.5f * qw, qy0 = qcy - 0.5f * qh;
  float qx1 = qcx + 0.5f * qw, qy1 = qcy + 0.5f * qh;
  float qarea = (qx1 - qx0) * (qy1 - qy0);
  float qsx = qx0 + qx1, qsy = qy0 + qy1;  // 2*cx, 2*cy

  // ---- Main loop: lanes stride targets -> coalesced 128B stores per wave.
  float* orow = out + (size_t)q * NT;
#pragma unroll 2
  for (int j = lane; j < NT; j += 32) {
    float4 t  = tb4[j];                      // ds_load_b128 (xyxy)
    float2 ax = taux[j];                     // ds_load_b64 (area, label)
    float  ta = ax.x;
    int    lb = __float_as_int(ax.y);
    float  p  = probs[wave][lb];             // class-prob gather (LDS)

    // L1 in cxcywh space, derived from xyxy: |dcx|+|dcy| = 0.5*|dsum|
    float bw = t.z - t.x, bh = t.w - t.y;
    float l1 = 0.5f * (fabsf(qsx - (t.x + t.z)) + fabsf(qsy - (t.y + t.w)))
             + fabsf(qw - bw) + fabsf(qh - bh);

    // GIoU
    float iw = fminf(qx1, t.z) - fmaxf(qx0, t.x);
    float ih = fminf(qy1, t.w) - fmaxf(qy0, t.y);
    float inter = fmaxf(iw, 0.0f) * fmaxf(ih, 0.0f);
    float uni = qarea + ta - inter;
    float iou = inter * __builtin_amdgcn_rcpf(uni);
    float ew = fmaxf(qx1, t.z) - fminf(qx0, t.x);
    float eh = fmaxf(qy1, t.w) - fminf(qy0, t.y);
    float ea = fmaxf(ew, 0.0f) * fmaxf(eh, 0.0f);
    float giou = iou - (ea - uni) * __builtin_amdgcn_rcpf(ea);

    // C = 5*L1 + 1*(-p) + 2*(-giou); stream store (write-once output)
    __builtin_nontemporal_store(5.0f * l1 - p - 2.0f * giou, &orow[j]);
  }
}

extern "C" void kernel_launch(void* const* d_in, const int* in_sizes, int n_in,
                              void* d_out, int out_size, void* d_ws, size_t ws_size,
                              hipStream_t stream) {
  const float* logits  = (const float*)d_in[0];  // pred_logits [16,1000,91]
  const float* pboxes  = (const float*)d_in[1];  // pred_boxes  [16,1000,4]
  const float* tboxes  = (const float*)d_in[2];  // tgt_boxes   [1600,4]
  const int*   tlabels = (const int*)d_in[3];    // tgt_labels  [1600]
  float* out = (float*)d_out;

  int bq = in_sizes[1] / 4;                 // 16000 query rows
  int blocks = (bq + QPB - 1) / QPB;        // 2000 blocks of 8 waves
  detr_cost_kernel<<<blocks, TPB, 0, stream>>>(logits, pboxes, tboxes, tlabels,
                                               out, bq);
}